// CropUpsample_2834678415683
// MI455X (gfx1250) — compile-verified
//
#include <hip/hip_runtime.h>
#include <cstdint>
#include <cstddef>

// ---------------------------------------------------------------------------
// CropUpsample for MI455X (gfx1250, wave32)
//   X:    (16, 32, 512, 512) f32
//   mask: (16,  1, 512, 512) f32
//   out:  (16, 32, 256, 256) f32
// Bandwidth-bound bilinear crop-resize:
//   kernel 1: bbox reduction (LDS flags + ds atomics) + index/weight tables
//   kernel 2: async-stage source-row spans to LDS (ASYNCcnt path), gather
//             bilinear taps from LDS, coalesced stores.
// ---------------------------------------------------------------------------

#define N_BATCH 16
#define CCH     32
#define HIN     512
#define WIN     512
#define HOUT    256
#define WOUT    256
#define THRESH  0.5f
#define CPB     8                 // channels per block in gather kernel

// Per-batch table layout in d_ws (ints / float-bits):
// [ x0[256] | x1[256] | wx[256] | y0[256] | y1[256] | wy[256] ]
#define TB_STRIDE (6 * WOUT)

__global__ __launch_bounds__(1024)
void bbox_tables_kernel(const float* __restrict__ mask, int* __restrict__ tabs) {
    __shared__ unsigned s_col[WIN];   // column has any mask >= THRESH
    __shared__ unsigned s_row[HIN];   // row has any mask >= THRESH
    __shared__ int s_xmin, s_xmax, s_ymin, s_ymax;
    __shared__ int s_oy0, s_ox0;
    __shared__ float s_hc, s_wc;

    const int n   = blockIdx.x;
    const int tid = threadIdx.x;

    if (tid < WIN) { s_col[tid] = 0u; s_row[tid] = 0u; }
    if (tid == 0) { s_xmin = 0x7FFFFFFF; s_xmax = 0; s_ymin = 0x7FFFFFFF; s_ymax = 0; }
    __syncthreads();

    // Threshold scan of the (H,W) mask plane; idempotent LDS flag stores.
    const float* mp = mask + (size_t)n * (HIN * WIN);
    for (int i = tid; i < HIN * WIN; i += 1024) {
        if (mp[i] >= THRESH) {
            s_col[i & (WIN - 1)] = 1u;
            s_row[i >> 9]        = 1u;
        }
    }
    __syncthreads();

    // First/last set index via LDS atomics (threads 0..511: cols, 512..1023: rows).
    if (tid < WIN) {
        if (s_col[tid]) { atomicMin(&s_xmin, tid); atomicMax(&s_xmax, tid + 1); }
    } else {
        int y = tid - WIN;
        if (s_row[y]) { atomicMin(&s_ymin, y); atomicMax(&s_ymax, y + 1); }
    }
    __syncthreads();

    if (tid == 0) {
        // any_ == false  ->  lo = 0, hi = size  (matches reference lohi()).
        int xmin = s_xmax ? s_xmin : 0;
        int xmax = s_xmax ? s_xmax : WIN;
        int ymin = s_ymax ? s_ymin : 0;
        int ymax = s_ymax ? s_ymax : HIN;

        // _pad with L = HOUT/WOUT = 1.0  =>  HW = [w, h] exactly (round no-op).
        int h = ymax - ymin, w = xmax - xmin;
        int resty = max(w - h, 0);
        int restx = max(h - w, 0);
        int r1y = min(resty >> 1, ymin);
        int r1x = min(restx >> 1, xmin);
        int r2y = resty - r1y;
        int r2x = restx - r1x;
        int oy0 = ymin - r1y;
        int ox0 = xmin - r1x;
        int oy1 = min(ymax + r2y, HIN);   // _crop_resize clamps max side only
        int ox1 = min(xmax + r2x, WIN);

        s_oy0 = oy0; s_ox0 = ox0;
        s_hc = (float)(oy1 - oy0);
        s_wc = (float)(ox1 - ox0);
    }
    __syncthreads();

    // 256 threads emit the bilinear source tables, matching JAX f32 op order
    // (no FMA contraction: _rn intrinsics keep separate mul/add rounding).
    if (tid < WOUT) {
        int* tb = tabs + n * TB_STRIDE;
        const float o = (float)tid + 0.5f;
        const float hc = s_hc, wc = s_wc;
        {   // x tables
            float scale = __fdiv_rn(wc, (float)WOUT);
            float s  = fmaxf(__fadd_rn(__fmul_rn(o, scale), -0.5f), 0.0f);
            float fl = floorf(s);
            int i0 = (int)fl;
            int i1 = min(i0 + 1, (int)wc - 1);
            tb[tid]            = i0 + s_ox0;
            tb[WOUT + tid]     = i1 + s_ox0;
            tb[2 * WOUT + tid] = __float_as_int(s - fl);
        }
        {   // y tables
            float scale = __fdiv_rn(hc, (float)HOUT);
            float s  = fmaxf(__fadd_rn(__fmul_rn(o, scale), -0.5f), 0.0f);
            float fl = floorf(s);
            int i0 = (int)fl;
            int i1 = min(i0 + 1, (int)hc - 1);
            tb[3 * WOUT + tid] = i0 + s_oy0;
            tb[4 * WOUT + tid] = i1 + s_oy0;
            tb[5 * WOUT + tid] = __float_as_int(s - fl);
        }
    }
}

// Async-stage one 32-bit element from global into LDS (gfx1250 ASYNCcnt path).
// lds_off: LDS byte offset (low 32 bits of a flat shared pointer per ISA §10.2);
// gaddr:   64-bit global address in a VGPR pair.
__device__ __forceinline__ void async_load_b32_to_lds(uint32_t lds_off, uint64_t gaddr) {
    asm volatile("global_load_async_to_lds_b32 %0, %1, off"
                 :: "v"(lds_off), "v"(gaddr) : "memory");
}

__global__ __launch_bounds__(256)
void crop_resize_kernel(const float* __restrict__ X,
                        const int* __restrict__ tabs,
                        float* __restrict__ out) {
    __shared__ float s_r0[WIN];   // staged source row y0 span
    __shared__ float s_r1[WIN];   // staged source row y1 span

    const int ox = threadIdx.x;      // output column
    const int oy = blockIdx.x;       // output row
    const int cg = blockIdx.y;       // channel group (CPB channels)
    const int n  = blockIdx.z;       // batch

    const int* tb = tabs + n * TB_STRIDE;
    const int   x0   = tb[ox];
    const int   x1   = tb[WOUT + ox];
    const float wxv  = __int_as_float(tb[2 * WOUT + ox]);
    const int   y0   = tb[3 * WOUT + oy];
    const int   y1   = tb[4 * WOUT + oy];
    const float wyv  = __int_as_float(tb[5 * WOUT + oy]);
    const int   xlo  = tb[0];              // min source column (x0 is nondecreasing)
    const int   xhi  = tb[2 * WOUT - 1];   // max source column (x1[255])
    const int   span = xhi - xlo + 1;      // <= 512 == WIN

    const float cwx = 1.0f - wxv;
    const float cwy = 1.0f - wyv;
    const int dx0 = x0 - xlo;
    const int dx1 = x1 - xlo;

    const int c0 = cg * CPB;
    const float* plane = X + ((size_t)n * CCH + c0) * (size_t)(HIN * WIN);
    float* op = out + (((size_t)n * CCH + c0) * HOUT + oy) * (size_t)WOUT + ox;

    const size_t r0off = (size_t)y0 * WIN + (size_t)xlo;
    const size_t r1off = (size_t)y1 * WIN + (size_t)xlo;

    for (int c = 0; c < CPB; ++c) {
        const float* p = plane + (size_t)c * (HIN * WIN);
        // Stage both source-row spans into LDS with async copies: each needed
        // dword is fetched from global exactly once, fully coalesced.
        for (int i = ox; i < span; i += WOUT) {      // <= 2 iterations
            async_load_b32_to_lds((uint32_t)(uintptr_t)&s_r0[i],
                                  (uint64_t)(uintptr_t)(p + r0off + i));
            async_load_b32_to_lds((uint32_t)(uintptr_t)&s_r1[i],
                                  (uint64_t)(uintptr_t)(p + r1off + i));
        }
        if (c + 1 < CPB) {
            // Warm L2 for the next channel's rows (global_prefetch_b8).
            __builtin_prefetch(p + (HIN * WIN) + r0off, 0, 0);
            __builtin_prefetch(p + (HIN * WIN) + r1off, 0, 0);
        }
        // Each wave drains its own async copies, then the block-wide barrier
        // makes all staged data visible to every wave.
        asm volatile("s_wait_asynccnt 0x0" ::: "memory");
        __syncthreads();

        // Irregular bilinear taps now hit LDS (stride <= 2 -> <= 2-way conflicts).
        const float a  = s_r0[dx0];
        const float b  = s_r0[dx1];
        const float c2 = s_r1[dx0];
        const float d  = s_r1[dx1];
        // Keep the reference's association: (1-wy)*((1-wx)a+wx*b) + wy*(...)
        const float top = cwx * a  + wxv * b;
        const float bot = cwx * c2 + wxv * d;
        op[(size_t)c * (HOUT * WOUT)] = cwy * top + wyv * bot;

        __syncthreads();   // protect LDS before the next channel overwrites it
    }
}

extern "C" void kernel_launch(void* const* d_in, const int* in_sizes, int n_in,
                              void* d_out, int out_size, void* d_ws, size_t ws_size,
                              hipStream_t stream) {
    const float* X    = (const float*)d_in[0];   // (16,32,512,512) f32
    const float* mask = (const float*)d_in[1];   // (16,1,512,512)  f32
    int*   tabs = (int*)d_ws;                    // 16 * 6 * 256 ints = 96 KB
    float* out  = (float*)d_out;                 // (16,32,256,256) f32
    (void)in_sizes; (void)n_in; (void)out_size; (void)ws_size;

    bbox_tables_kernel<<<N_BATCH, 1024, 0, stream>>>(mask, tabs);
    crop_resize_kernel<<<dim3(HOUT, CCH / CPB, N_BATCH), WOUT, 0, stream>>>(X, tabs, out);
}